// LGCN_Encoder_57303453663962
// MI455X (gfx1250) — compile-verified
//
#include <hip/hip_runtime.h>
#include <cstdint>
#include <cstddef>

#define USER_NUM 50000
#define ITEM_NUM 50000
#define N_NODES  100000
#define EMB      64
#define N_LAYERS 3
#define ADJ_NNZ  1600000
#define S_NNZ    800000

#define TPB   256           // 8 waves per block (wave32)
#define WPB   (TPB / 32)
#define CHUNK 32            // nnz staged to LDS per wave per step

#if defined(__HIP_DEVICE_COMPILE__) && __has_builtin(__builtin_amdgcn_global_load_async_to_lds_b32)
#define HAS_ASYNC 1
#else
#define HAS_ASYNC 0
#endif

#if HAS_ASYNC
#define AS1 __attribute__((address_space(1)))
#define AS3 __attribute__((address_space(3)))
// Probe-confirmed: param0 is a (non-const) addrspace(1) int*.
__device__ __forceinline__ void async_g2l_b32(const void* g, void* l) {
  __builtin_amdgcn_global_load_async_to_lds_b32((AS1 int*)g, (AS3 int*)l, 0, 0);
}
#endif

// ---------------- init: ego = concat(user,item); acc(d_out) = ego -------------
__global__ __launch_bounds__(TPB) void k_init(const float* __restrict__ ue,
                                              const float* __restrict__ ie,
                                              float* __restrict__ ego,
                                              float* __restrict__ acc) {
  int i = blockIdx.x * TPB + threadIdx.x;
  if (i < N_NODES * EMB) {
    float v = (i < USER_NUM * EMB) ? ue[i] : ie[i - USER_NUM * EMB];
    ego[i] = v;
    acc[i] = v;
  }
}

__global__ __launch_bounds__(TPB) void k_zero(int* __restrict__ p, int n) {
  int i = blockIdx.x * TPB + threadIdx.x;
  if (i < n) p[i] = 0;
}

// ---------------- CSR build: count / scan / scatter ---------------------------
__global__ __launch_bounds__(TPB) void k_count(const int* __restrict__ rows, int nnz,
                                               int* __restrict__ rowptr) {
  int i = blockIdx.x * TPB + threadIdx.x;
  if (i < nnz) atomicAdd(&rowptr[rows[i] + 1], 1);
}

__global__ __launch_bounds__(1024) void k_scan(int* __restrict__ rowptr,
                                               int* __restrict__ pos, int n) {
  __shared__ int part[1024];
  const int t = threadIdx.x;
  const int T = blockDim.x;
  const int chunk = (n + T - 1) / T;
  const int begin = t * chunk;
  const int end   = (begin + chunk < n) ? (begin + chunk) : n;
  int s = 0;
  for (int r = begin; r < end; ++r) s += rowptr[r + 1];
  part[t] = s;
  __syncthreads();
  for (int off = 1; off < T; off <<= 1) {
    int v = part[t];
    int add = (t >= off) ? part[t - off] : 0;
    __syncthreads();
    part[t] = v + add;
    __syncthreads();
  }
  int P = (t == 0) ? 0 : part[t - 1];
  for (int r = begin; r < end; ++r) {
    int c = rowptr[r + 1];
    pos[r] = P;
    P += c;
    rowptr[r + 1] = P;
  }
  if (t == 0) rowptr[0] = 0;
}

__global__ __launch_bounds__(TPB) void k_scatter(const int* __restrict__ rows,
                                                 const int* __restrict__ cols,
                                                 const float* __restrict__ vals, int nnz,
                                                 int* __restrict__ pos,
                                                 int* __restrict__ cols_s,
                                                 float* __restrict__ vals_s) {
  int i = blockIdx.x * TPB + threadIdx.x;
  if (i < nnz) {
    int p = atomicAdd(&pos[rows[i]], 1);
    cols_s[p] = cols[i];
    vals_s[p] = vals[i];
  }
}

// ---------------- CSR SpMM: one wave32 per row, float2 per lane ---------------
// gather(c) = c < nsplit ? xu[c] : xv[c - nsplit]
// ADD_SRC: y[row] = src[row] + A x   (for u + S.u)
// ACC:     acc[row] += y[row]        (fused running sum for the layer output)
template <bool ADD_SRC, bool ACC>
__global__ __launch_bounds__(TPB) void k_spmm(const int* __restrict__ rowptr,
                                              const int* __restrict__ cols,
                                              const float* __restrict__ vals,
                                              const float* __restrict__ xu,
                                              const float* __restrict__ xv, int nsplit,
                                              const float* __restrict__ src,
                                              float* __restrict__ y,
                                              float* __restrict__ acc, int nrows) {
  // (col,val) interleaved: col at slot 2j, val at slot 2j+1 -> one ds_load_b64/nnz
  __shared__ int sh[WPB][2 * CHUNK];
  const int lane = threadIdx.x & 31;
  const int wid  = threadIdx.x >> 5;
  const int row  = blockIdx.x * WPB + wid;
  if (row >= nrows) return;

  const int start = rowptr[row];
  const int end   = rowptr[row + 1];
  float a0 = 0.f, a1 = 0.f;

  for (int base = start; base < end; base += CHUNK) {
    const int m = (end - base < CHUNK) ? (end - base) : CHUNK;
#if HAS_ASYNC
    asm volatile("s_wait_dscnt 0" ::: "memory");  // WAR vs prior chunk's LDS reads
    if (lane < m) {
      async_g2l_b32(cols + base + lane, &sh[wid][2 * lane]);
      async_g2l_b32(vals + base + lane, &sh[wid][2 * lane + 1]);
    }
#if __has_builtin(__builtin_amdgcn_s_wait_asynccnt)
    __builtin_amdgcn_s_wait_asynccnt(0);
#else
    asm volatile("s_wait_asynccnt 0" ::: "memory");
#endif
#else
    if (lane < m) {
      sh[wid][2 * lane]     = cols[base + lane];
      sh[wid][2 * lane + 1] = __float_as_int(vals[base + lane]);
    }
#endif
    __builtin_amdgcn_wave_barrier();

    // 4-wide software pipeline: 4 gathers in flight before any use
    int j = 0;
    for (; j + 4 <= m; j += 4) {
      const int2 cw0 = *(const int2*)&sh[wid][2 * (j + 0)];
      const int2 cw1 = *(const int2*)&sh[wid][2 * (j + 1)];
      const int2 cw2 = *(const int2*)&sh[wid][2 * (j + 2)];
      const int2 cw3 = *(const int2*)&sh[wid][2 * (j + 3)];
      const float* b0 = (cw0.x < nsplit) ? (xu + (size_t)cw0.x * EMB)
                                         : (xv + (size_t)(cw0.x - nsplit) * EMB);
      const float* b1 = (cw1.x < nsplit) ? (xu + (size_t)cw1.x * EMB)
                                         : (xv + (size_t)(cw1.x - nsplit) * EMB);
      const float* b2 = (cw2.x < nsplit) ? (xu + (size_t)cw2.x * EMB)
                                         : (xv + (size_t)(cw2.x - nsplit) * EMB);
      const float* b3 = (cw3.x < nsplit) ? (xu + (size_t)cw3.x * EMB)
                                         : (xv + (size_t)(cw3.x - nsplit) * EMB);
      const float2 x0 = *(const float2*)(b0 + lane * 2);
      const float2 x1 = *(const float2*)(b1 + lane * 2);
      const float2 x2 = *(const float2*)(b2 + lane * 2);
      const float2 x3 = *(const float2*)(b3 + lane * 2);
      const float w0 = __int_as_float(cw0.y);
      const float w1 = __int_as_float(cw1.y);
      const float w2 = __int_as_float(cw2.y);
      const float w3 = __int_as_float(cw3.y);
      a0 = fmaf(w0, x0.x, a0); a1 = fmaf(w0, x0.y, a1);
      a0 = fmaf(w1, x1.x, a0); a1 = fmaf(w1, x1.y, a1);
      a0 = fmaf(w2, x2.x, a0); a1 = fmaf(w2, x2.y, a1);
      a0 = fmaf(w3, x3.x, a0); a1 = fmaf(w3, x3.y, a1);
    }
    for (; j < m; ++j) {
      const int2 cw = *(const int2*)&sh[wid][2 * j];
      const float* b = (cw.x < nsplit) ? (xu + (size_t)cw.x * EMB)
                                       : (xv + (size_t)(cw.x - nsplit) * EMB);
      const float2 xw = *(const float2*)(b + lane * 2);
      const float w = __int_as_float(cw.y);
      a0 = fmaf(w, xw.x, a0);
      a1 = fmaf(w, xw.y, a1);
    }
    __builtin_amdgcn_wave_barrier();
  }

  if (ADD_SRC) {
    const float2 s = *(const float2*)(src + (size_t)row * EMB + lane * 2);
    a0 += s.x;
    a1 += s.y;
  }
  float2 o;
  o.x = a0;
  o.y = a1;
  *(float2*)(y + (size_t)row * EMB + lane * 2) = o;
  if (ACC) {
    float* ap = acc + (size_t)row * EMB + lane * 2;
    ap[0] += a0;
    ap[1] += a1;
  }
}

__global__ __launch_bounds__(TPB) void k_scale(float* __restrict__ out, int n) {
  int i = blockIdx.x * TPB + threadIdx.x;
  if (i < n) out[i] *= 0.25f;
}

// ------------------------------- launch ---------------------------------------
static inline size_t align256(size_t x) { return (x + 255) & ~(size_t)255; }

extern "C" void kernel_launch(void* const* d_in, const int* in_sizes, int n_in,
                              void* d_out, int out_size, void* d_ws, size_t ws_size,
                              hipStream_t stream) {
  const float* user_emb = (const float*)d_in[0];
  const float* item_emb = (const float*)d_in[1];
  const int*   adj_rows = (const int*)d_in[2];
  const int*   adj_cols = (const int*)d_in[3];
  const float* adj_vals = (const float*)d_in[4];
  const int*   s_rows   = (const int*)d_in[5];
  const int*   s_cols   = (const int*)d_in[6];
  const float* s_vals   = (const float*)d_in[7];
  float* out = (float*)d_out;  // acc buffer == output layout (user || item)

  char* w = (char*)d_ws;
  auto carve = [&](size_t bytes) {
    char* p = w;
    w += align256(bytes);
    return (void*)p;
  };
  float* ego0       = (float*)carve((size_t)N_NODES * EMB * 4);
  float* ego1       = (float*)carve((size_t)N_NODES * EMB * 4);
  float* ubuf       = (float*)carve((size_t)USER_NUM * EMB * 4);
  int*   adj_rowptr = (int*)carve((size_t)(N_NODES + 1) * 4);
  int*   adj_pos    = (int*)carve((size_t)N_NODES * 4);
  int*   adj_cols_s = (int*)carve((size_t)ADJ_NNZ * 4);
  float* adj_vals_s = (float*)carve((size_t)ADJ_NNZ * 4);
  int*   s_rowptr   = (int*)carve((size_t)(USER_NUM + 1) * 4);
  int*   s_pos      = (int*)carve((size_t)USER_NUM * 4);
  int*   s_cols_s   = (int*)carve((size_t)S_NNZ * 4);
  float* s_vals_s   = (float*)carve((size_t)S_NNZ * 4);

  const int total = N_NODES * EMB;
  k_init<<<(total + TPB - 1) / TPB, TPB, 0, stream>>>(user_emb, item_emb, ego0, out);

  // ---- build CSR for both matrices (once per launch, reused for 3 layers) ----
  k_zero<<<(N_NODES + 1 + TPB - 1) / TPB, TPB, 0, stream>>>(adj_rowptr, N_NODES + 1);
  k_zero<<<(USER_NUM + 1 + TPB - 1) / TPB, TPB, 0, stream>>>(s_rowptr, USER_NUM + 1);
  k_count<<<(ADJ_NNZ + TPB - 1) / TPB, TPB, 0, stream>>>(adj_rows, ADJ_NNZ, adj_rowptr);
  k_count<<<(S_NNZ + TPB - 1) / TPB, TPB, 0, stream>>>(s_rows, S_NNZ, s_rowptr);
  k_scan<<<1, 1024, 0, stream>>>(adj_rowptr, adj_pos, N_NODES);
  k_scan<<<1, 1024, 0, stream>>>(s_rowptr, s_pos, USER_NUM);
  k_scatter<<<(ADJ_NNZ + TPB - 1) / TPB, TPB, 0, stream>>>(adj_rows, adj_cols, adj_vals,
                                                           ADJ_NNZ, adj_pos, adj_cols_s,
                                                           adj_vals_s);
  k_scatter<<<(S_NNZ + TPB - 1) / TPB, TPB, 0, stream>>>(s_rows, s_cols, s_vals, S_NNZ,
                                                         s_pos, s_cols_s, s_vals_s);

  // ---- 3 propagation layers, atomic-free, acc fused into adj-SpMM ----
  float* ego = ego0;
  float* nxt = ego1;
  for (int l = 0; l < N_LAYERS; ++l) {
    // u' = u + S.u   (rows: USER_NUM; gather entirely from user slice of ego)
    k_spmm<true, false><<<(USER_NUM + WPB - 1) / WPB, TPB, 0, stream>>>(
        s_rowptr, s_cols_s, s_vals_s, ego, ego, USER_NUM, ego, ubuf, nullptr, USER_NUM);
    // ego' = A . concat(u', v); acc += ego'
    k_spmm<false, true><<<(N_NODES + WPB - 1) / WPB, TPB, 0, stream>>>(
        adj_rowptr, adj_cols_s, adj_vals_s, ubuf, ego + (size_t)USER_NUM * EMB, USER_NUM,
        nullptr, nxt, out, N_NODES);
    float* t = ego;
    ego = nxt;
    nxt = t;
  }

  k_scale<<<(total + TPB - 1) / TPB, TPB, 0, stream>>>(out, total);
}